// FixedPointIterationFast_88175678587248
// MI455X (gfx1250) — compile-verified
//
#include <hip/hip_runtime.h>
#include <stdint.h>

// ---------------------------------------------------------------------------
// Gamma-smoothed forward-backward on independent chains (CRF mean-field step).
// B=4, K=21, H=W=256.
//   * memory-bound: ~2.1GB moved -> ~90us floor at 23.3 TB/s.
//   * horizontal chains: steps are the contiguous dim -> TDM tensor_load_to_lds
//     stages an 8-step x 441-row strided tile into LDS (double buffered,
//     s_wait_tensorcnt), pad_enable gives 9-dword rows (bank-conflict free).
//   * vertical chains: chains are the contiguous dim -> direct coalesced loads,
//     4 chains x 8 state-groups per wave32, shfl_xor reductions.
//   * exact normalized linear-space recursion (scales cancel in softmax),
//     so no logs; one v_exp per pairwise element.
// ---------------------------------------------------------------------------

#define KST 21
#define BB  4
#define HH  256
#define WW  256
#define HWSZ (HH * WW)                 // 65536
#define QS  (KST * KST)                // 441 (k,k') streams
#define QSTRIDE 65280ull               // H*(W-1) == (H-1)*W, elements
#define NEDGE 255
#define SPB 8                          // steps per TDM tile
#define NBLK 32                        // ceil(255/8)
#define ROWPAD 9                       // 8 data dwords + 1 pad dword per LDS row
#define EBUF (QS * ROWPAD)             // 3969 dwords per buffer
#define NEG_INF (-3.402823466e38f)

typedef unsigned int uint32x4 __attribute__((ext_vector_type(4)));
typedef int int32x8 __attribute__((ext_vector_type(8)));
typedef int int32x4 __attribute__((ext_vector_type(4)));

// Issue one TDM load: 441 rows x 8 floats, row stride 65280 floats, with LDS
// padding of 1 dword after every 8 dwords (row stride 9 in LDS).
static __device__ __forceinline__ void tdm_issue_tile(const float* gsrc, unsigned lds_off) {
  unsigned long long ga = (unsigned long long)(uintptr_t)gsrc;
  uint32x4 g0;
  g0.x = 1u;                                              // count=1, user descriptor
  g0.y = lds_off;                                         // LDS byte address
  g0.z = (unsigned)(ga & 0xFFFFFFFFull);                  // global_addr[31:0]
  g0.w = (unsigned)((ga >> 32) & 0x01FFFFFFull) | (2u << 30); // addr[56:32], type=2
  int32x8 g1;
  g1[0] = (int)((2u << 16) | (1u << 20) | (2u << 22));    // data_size=4B, pad_en, pad every 8dw (+1dw)
  g1[1] = (int)0xFFFF0000u;                               // tensor_dim0 lo16 (huge, no clip)
  g1[2] = (int)(0x0FFFu | (0xFFFFu << 16));               // tensor_dim0 hi16, tensor_dim1 lo16
  g1[3] = (int)(0x0FFFu | (8u << 16));                    // tensor_dim1 hi16, tile_dim0=8
  g1[4] = (int)QS;                                        // tile_dim1=441, tile_dim2=0
  g1[5] = (int)65280;                                     // tensor_dim0_stride (elements)
  g1[6] = 0;
  g1[7] = 0;
  int32x4 z4; z4[0] = 0; z4[1] = 0; z4[2] = 0; z4[3] = 0; // 2D tensor: groups 2/3 unused
  int32x8 z8; z8[0] = 0; z8[1] = 0; z8[2] = 0; z8[3] = 0;
  z8[4] = 0; z8[5] = 0; z8[6] = 0; z8[7] = 0;
  __builtin_amdgcn_tensor_load_to_lds(g0, g1, z4, z4, z8, 0);
}

static __device__ __forceinline__ float wsum32(float v) {
#pragma unroll
  for (int m = 16; m >= 1; m >>= 1) v += __shfl_xor(v, m, 32);
  return v;
}
static __device__ __forceinline__ float wmax32(float v) {
#pragma unroll
  for (int m = 16; m >= 1; m >>= 1) v = fmaxf(v, __shfl_xor(v, m, 32));
  return v;
}
// reduce across the 8 state-groups (lane bits [4:2]); lanes sharing bits [1:0]
// (the chain id) combine.
static __device__ __forceinline__ float gsum8(float v) {
  v += __shfl_xor(v, 4, 32); v += __shfl_xor(v, 8, 32); v += __shfl_xor(v, 16, 32);
  return v;
}
static __device__ __forceinline__ float gmax8(float v) {
  v = fmaxf(v, __shfl_xor(v, 4, 32));
  v = fmaxf(v, __shfl_xor(v, 8, 32));
  v = fmaxf(v, __shfl_xor(v, 16, 32));
  return v;
}

// ===========================================================================
// Backward pass: writes normalized Bt (exp(beta/g) normalized) for both
// directions. btH -> new_uh slot, btV -> new_uv slot (scratch until merge).
// ===========================================================================
__global__ __launch_bounds__(32) void fb_backward(
    const float* __restrict__ uh, const float* __restrict__ uv,
    const float* __restrict__ ph, const float* __restrict__ pv,
    const float* __restrict__ gm,
    float* __restrict__ btH, float* __restrict__ btV) {
  __shared__ float sE[2][EBUF];
  __shared__ float sX[4][KST];
  const int lane = threadIdx.x;
  const float g  = fmaxf(gm[0], 0.01f);
  const float ig = 1.0f / g;
  const int bid  = blockIdx.x;

  if (bid < BB * HH) {
    // ----- horizontal chain (b,h): one wave, lane = state, TDM-staged E -----
    const int b = bid >> 8, h = bid & (HH - 1);
    const bool act = lane < KST;
    const int l = act ? lane : 0;
    const float* up    = uh + (size_t)(b * KST + l) * HWSZ + (size_t)h * WW;
    const float* pbase = ph + (size_t)b * QS * QSTRIDE + (size_t)h * NEDGE;
    float* bto = btH + (size_t)bid * WW * KST;          // [chain][i][K]
    float bt = 1.0f / KST;
    if (act) bto[(size_t)(WW - 1) * KST + lane] = bt;   // Bt[L-1] uniform
    const unsigned off0 = (unsigned)(uintptr_t)&sE[0][0];
    const unsigned off1 = (unsigned)(uintptr_t)&sE[1][0];
    tdm_issue_tile(pbase + SPB * (NBLK - 1), off0);
    tdm_issue_tile(pbase + SPB * (NBLK - 2), off1);
    for (int j = NBLK - 1; j >= 0; --j) {
      if (j > 0) __builtin_amdgcn_s_wait_tensorcnt(1);
      else       __builtin_amdgcn_s_wait_tensorcnt(0);
      const float* eb = &sE[(NBLK - 1 - j) & 1][0];
      const int elo = SPB * j;
      const int ehi = (j == NBLK - 1) ? (NEDGE - 1) : (elo + SPB - 1);
      for (int e = ehi; e >= elo; --e) {
        const int s = e - elo;
        float ul = act ? up[e + 1] : NEG_INF;
        float um = wmax32(ul);
        float wv = act ? __expf((ul - um) * ig) * bt : 0.0f;
        float acc = 0.0f;
#pragma unroll
        for (int kp = 0; kp < KST; ++kp) {
          float wk = __shfl(wv, kp, 32);
          acc = fmaf(wk, __expf(eb[(l * KST + kp) * ROWPAD + s] * ig), acc);
        }
        float ns = wsum32(act ? acc : 0.0f);
        bt = acc * (1.0f / ns);
        if (act) bto[(size_t)e * KST + lane] = bt;
      }
      if (j >= 2) tdm_issue_tile(pbase + SPB * (j - 2), ((NBLK - 1 - j) & 1) ? off1 : off0);
    }
  } else {
    // ----- vertical: 4 chains (lane bits[1:0]) x 8 state-groups per wave -----
    const int vcb = bid - BB * HH;        // 0..255 (4 chains each)
    const int b  = vcb >> 6;
    const int w0 = (vcb & 63) * 4;
    const int c  = lane & 3;
    const int grp = lane >> 2;
    const int kb = (grp < 5) ? 3 * grp : 15 + 2 * (grp - 5);
    const int kc = (grp < 5) ? 3 : 2;
    float* bto = btV + (size_t)vcb * HH * (KST * 4);    // [(vcb*H+i)*84 + k*4 + c]
    float bt0 = 1.0f / KST, bt1 = bt0, bt2 = bt0;
    {
      size_t ob = ((size_t)vcb * HH + (HH - 1)) * (KST * 4);
      bto[ob + (size_t)kb * 4 + c] = bt0;
      if (1 < kc) bto[ob + (size_t)(kb + 1) * 4 + c] = bt1;
      if (2 < kc) bto[ob + (size_t)(kb + 2) * 4 + c] = bt2;
    }
    for (int i = HH - 2; i >= 0; --i) {
      size_t ub = ((size_t)(b * KST + kb) * HH + (i + 1)) * WW + w0 + c;
      float u0 = uv[ub];
      float u1 = (1 < kc) ? uv[ub + (size_t)HWSZ] : NEG_INF;
      float u2 = (2 < kc) ? uv[ub + 2 * (size_t)HWSZ] : NEG_INF;
      float um = gmax8(fmaxf(u0, fmaxf(u1, u2)));
      float w_0 = __expf((u0 - um) * ig) * bt0;
      float w_1 = (1 < kc) ? __expf((u1 - um) * ig) * bt1 : 0.0f;
      float w_2 = (2 < kc) ? __expf((u2 - um) * ig) * bt2 : 0.0f;
      sX[c][kb] = w_0;
      if (1 < kc) sX[c][kb + 1] = w_1;
      if (2 < kc) sX[c][kb + 2] = w_2;
      __builtin_amdgcn_wave_barrier();
      float s0 = 0.0f, s1 = 0.0f, s2 = 0.0f;
      // P(b, k=kb+j, kp, i, w0+c); kp stride = QSTRIDE, k stride = K*QSTRIDE
      size_t pb0 = ((size_t)(b * KST + kb) * KST) * QSTRIDE + (size_t)i * WW + w0 + c;
#pragma unroll
      for (int kp = 0; kp < KST; ++kp) {
        float wk = sX[c][kp];
        size_t o = pb0 + (size_t)kp * QSTRIDE;
        s0 = fmaf(wk, __expf(pv[o] * ig), s0);
        if (1 < kc) s1 = fmaf(wk, __expf(pv[o + (size_t)KST * QSTRIDE] * ig), s1);
        if (2 < kc) s2 = fmaf(wk, __expf(pv[o + 2 * (size_t)KST * QSTRIDE] * ig), s2);
      }
      float ns = gsum8(s0 + s1 + s2);
      float r = 1.0f / ns;
      bt0 = s0 * r; bt1 = s1 * r; bt2 = s2 * r;
      size_t ob = ((size_t)vcb * HH + i) * (KST * 4);
      bto[ob + (size_t)kb * 4 + c] = bt0;
      if (1 < kc) bto[ob + (size_t)(kb + 1) * 4 + c] = bt1;
      if (2 < kc) bto[ob + (size_t)(kb + 2) * 4 + c] = bt2;
      __builtin_amdgcn_wave_barrier();
    }
  }
}

// ===========================================================================
// Forward pass: streams alpha, reads Bt, emits normalized marginals mh / mv.
// ===========================================================================
__global__ __launch_bounds__(32) void fb_forward(
    const float* __restrict__ uh, const float* __restrict__ uv,
    const float* __restrict__ ph, const float* __restrict__ pv,
    const float* __restrict__ gm,
    const float* __restrict__ btH, const float* __restrict__ btV,
    float* __restrict__ mh, float* __restrict__ mv) {
  __shared__ float sE[2][EBUF];
  __shared__ float sX[4][KST];
  const int lane = threadIdx.x;
  const float g  = fmaxf(gm[0], 0.01f);
  const float ig = 1.0f / g;
  const int bid  = blockIdx.x;

  if (bid < BB * HH) {
    // ----- horizontal forward -----
    const int b = bid >> 8, h = bid & (HH - 1);
    const bool act = lane < KST;
    const int l = act ? lane : 0;
    const float* up    = uh + (size_t)(b * KST + l) * HWSZ + (size_t)h * WW;
    const float* pbase = ph + (size_t)b * QS * QSTRIDE + (size_t)h * NEDGE;
    const float* bti   = btH + (size_t)bid * WW * KST;
    float* mho = mh + (size_t)(b * KST + l) * HWSZ + (size_t)h * WW;
    float x;
    { // i = 0
      float ul = act ? up[0] : NEG_INF;
      float um = wmax32(ul);
      float ev = act ? __expf((ul - um) * ig) : 0.0f;
      float ns = wsum32(ev);
      x = ev * (1.0f / ns);
      float bv = act ? bti[l] : 0.0f;
      float mn = x * bv;
      float ms = wsum32(mn);
      if (act) mho[0] = mn * (1.0f / ms);
    }
    const unsigned off0 = (unsigned)(uintptr_t)&sE[0][0];
    const unsigned off1 = (unsigned)(uintptr_t)&sE[1][0];
    tdm_issue_tile(pbase, off0);
    tdm_issue_tile(pbase + SPB, off1);
    for (int j = 0; j < NBLK; ++j) {
      if (j < NBLK - 1) __builtin_amdgcn_s_wait_tensorcnt(1);
      else              __builtin_amdgcn_s_wait_tensorcnt(0);
      const float* eb = &sE[j & 1][0];
      const int elo = SPB * j;
      const int ehi = (j == NBLK - 1) ? (NEDGE - 1) : (elo + SPB - 1);
      for (int e = elo; e <= ehi; ++e) {
        const int i = e + 1, s = e - elo;
        float ul = act ? up[i] : NEG_INF;
        float um = wmax32(ul);
        float ue = act ? __expf((ul - um) * ig) : 0.0f;
        float acc = 0.0f;
#pragma unroll
        for (int k = 0; k < KST; ++k) {
          float xk = __shfl(x, k, 32);
          acc = fmaf(xk, __expf(eb[(k * KST + l) * ROWPAD + s] * ig), acc);
        }
        float y = act ? ue * acc : 0.0f;
        float ns = wsum32(y);
        x = y * (1.0f / ns);
        float bv = act ? bti[(size_t)i * KST + l] : 0.0f;
        float mn = x * bv;
        float ms = wsum32(mn);
        if (act) mho[i] = mn * (1.0f / ms);
      }
      if (j + 2 < NBLK) tdm_issue_tile(pbase + SPB * (j + 2), (j & 1) ? off1 : off0);
    }
  } else {
    // ----- vertical forward -----
    const int vcb = bid - BB * HH;
    const int b  = vcb >> 6;
    const int w0 = (vcb & 63) * 4;
    const int c  = lane & 3;
    const int grp = lane >> 2;
    const int kb = (grp < 5) ? 3 * grp : 15 + 2 * (grp - 5);
    const int kc = (grp < 5) ? 3 : 2;
    const float* bti = btV + (size_t)vcb * HH * (KST * 4);
    float a0, a1 = 0.0f, a2 = 0.0f;
    { // i = 0
      size_t ub = ((size_t)(b * KST + kb) * HH) * WW + w0 + c;
      float u0 = uv[ub];
      float u1 = (1 < kc) ? uv[ub + (size_t)HWSZ] : NEG_INF;
      float u2 = (2 < kc) ? uv[ub + 2 * (size_t)HWSZ] : NEG_INF;
      float um = gmax8(fmaxf(u0, fmaxf(u1, u2)));
      float e0 = __expf((u0 - um) * ig);
      float e1 = (1 < kc) ? __expf((u1 - um) * ig) : 0.0f;
      float e2 = (2 < kc) ? __expf((u2 - um) * ig) : 0.0f;
      float r = 1.0f / gsum8(e0 + e1 + e2);
      a0 = e0 * r; a1 = e1 * r; a2 = e2 * r;
      size_t ob = 0;
      float b0 = bti[ob + (size_t)kb * 4 + c];
      float b1 = (1 < kc) ? bti[ob + (size_t)(kb + 1) * 4 + c] : 0.0f;
      float b2 = (2 < kc) ? bti[ob + (size_t)(kb + 2) * 4 + c] : 0.0f;
      float m0 = a0 * b0, m1 = a1 * b1, m2 = a2 * b2;
      float rm = 1.0f / gsum8(m0 + m1 + m2);
      size_t mo = ((size_t)(b * KST + kb) * HH) * WW + w0 + c;
      mv[mo] = m0 * rm;
      if (1 < kc) mv[mo + (size_t)HWSZ] = m1 * rm;
      if (2 < kc) mv[mo + 2 * (size_t)HWSZ] = m2 * rm;
    }
    for (int i = 1; i < HH; ++i) {
      sX[c][kb] = a0;
      if (1 < kc) sX[c][kb + 1] = a1;
      if (2 < kc) sX[c][kb + 2] = a2;
      __builtin_amdgcn_wave_barrier();
      // s[kp=kb+j] = sum_k A[k] * exp(P(b,k,kp,i-1,w0+c)/g)
      float s0 = 0.0f, s1 = 0.0f, s2 = 0.0f;
      size_t rowoff = (size_t)(i - 1) * WW + w0 + c;
#pragma unroll
      for (int k = 0; k < KST; ++k) {
        float xk = sX[c][k];
        size_t o = ((size_t)(b * KST + k) * KST + kb) * QSTRIDE + rowoff;
        s0 = fmaf(xk, __expf(pv[o] * ig), s0);
        if (1 < kc) s1 = fmaf(xk, __expf(pv[o + QSTRIDE] * ig), s1);
        if (2 < kc) s2 = fmaf(xk, __expf(pv[o + 2 * QSTRIDE] * ig), s2);
      }
      size_t ub = ((size_t)(b * KST + kb) * HH + i) * WW + w0 + c;
      float u0 = uv[ub];
      float u1 = (1 < kc) ? uv[ub + (size_t)HWSZ] : NEG_INF;
      float u2 = (2 < kc) ? uv[ub + 2 * (size_t)HWSZ] : NEG_INF;
      float um = gmax8(fmaxf(u0, fmaxf(u1, u2)));
      float y0 = __expf((u0 - um) * ig) * s0;
      float y1 = (1 < kc) ? __expf((u1 - um) * ig) * s1 : 0.0f;
      float y2 = (2 < kc) ? __expf((u2 - um) * ig) * s2 : 0.0f;
      float r = 1.0f / gsum8(y0 + y1 + y2);
      a0 = y0 * r; a1 = y1 * r; a2 = y2 * r;
      size_t ob = (size_t)i * (KST * 4);
      float b0 = bti[ob + (size_t)kb * 4 + c];
      float b1 = (1 < kc) ? bti[ob + (size_t)(kb + 1) * 4 + c] : 0.0f;
      float b2 = (2 < kc) ? bti[ob + (size_t)(kb + 2) * 4 + c] : 0.0f;
      float m0 = a0 * b0, m1 = a1 * b1, m2 = a2 * b2;
      float rm = 1.0f / gsum8(m0 + m1 + m2);
      size_t mo = ((size_t)(b * KST + kb) * HH + i) * WW + w0 + c;
      mv[mo] = m0 * rm;
      if (1 < kc) mv[mo + (size_t)HWSZ] = m1 * rm;
      if (2 < kc) mv[mo + 2 * (size_t)HWSZ] = m2 * rm;
      __builtin_amdgcn_wave_barrier();
    }
  }
}

// ===========================================================================
// Merge: new_uh = uh - (mh-mv)/512 ; new_uv = uv + (mh-mv)/512.
// Overwrites the Bt scratch living in the new_uh / new_uv slots.
// ===========================================================================
__global__ void fb_merge(const float4* __restrict__ uh4, const float4* __restrict__ uv4,
                         const float4* __restrict__ mh4, const float4* __restrict__ mv4,
                         float4* __restrict__ nuh4, float4* __restrict__ nuv4, int n4) {
  int i = blockIdx.x * blockDim.x + threadIdx.x;
  if (i >= n4) return;
  float4 a = mh4[i], bv = mv4[i], x = uh4[i], y = uv4[i];
  const float s = 1.0f / 512.0f;  // 1/(2*W) == 1/(2*H)
  float4 o1, o2;
  o1.x = x.x - (a.x - bv.x) * s; o1.y = x.y - (a.y - bv.y) * s;
  o1.z = x.z - (a.z - bv.z) * s; o1.w = x.w - (a.w - bv.w) * s;
  o2.x = y.x + (a.x - bv.x) * s; o2.y = y.y + (a.y - bv.y) * s;
  o2.z = y.z + (a.z - bv.z) * s; o2.w = y.w + (a.w - bv.w) * s;
  nuh4[i] = o1;
  nuv4[i] = o2;
}

extern "C" void kernel_launch(void* const* d_in, const int* in_sizes, int n_in,
                              void* d_out, int out_size, void* d_ws, size_t ws_size,
                              hipStream_t stream) {
  (void)in_sizes; (void)n_in; (void)out_size; (void)d_ws; (void)ws_size;
  const float* uh = (const float*)d_in[0];
  const float* uv = (const float*)d_in[1];
  const float* ph = (const float*)d_in[2];
  const float* pv = (const float*)d_in[3];
  const float* gm = (const float*)d_in[4];
  float* out = (float*)d_out;
  const size_t N = (size_t)BB * KST * HH * WW;   // 5,505,024 per output tensor
  float* new_uh = out;
  float* new_uv = out + N;
  float* mh = out + 2 * N;
  float* mv = out + 3 * N;

  const int nblocks = BB * HH + 256;   // 1024 horizontal chains + 256 vertical chain-quads
  fb_backward<<<nblocks, 32, 0, stream>>>(uh, uv, ph, pv, gm, new_uh, new_uv);
  fb_forward<<<nblocks, 32, 0, stream>>>(uh, uv, ph, pv, gm, new_uh, new_uv, mh, mv);

  const int n4 = (int)(N / 4);
  fb_merge<<<(n4 + 255) / 256, 256, 0, stream>>>(
      (const float4*)uh, (const float4*)uv, (const float4*)mh, (const float4*)mv,
      (float4*)new_uh, (float4*)new_uv, n4);
}